// SpGraphAttentionLayer_730144441124
// MI455X (gfx1250) — compile-verified
//
#include <hip/hip_runtime.h>
#include <hip/hip_bf16.h>
#include <math.h>

typedef __attribute__((ext_vector_type(2))) float v2f;
typedef __attribute__((ext_vector_type(8))) float v8f;

#define GAT_ALPHA 0.2f
#define IN_F 256
#define OUT_F 64

// ---------------------------------------------------------------------------
// Kernel 1: h = x @ W via V_WMMA_F32_16X16X4_F32 (full f32 precision).
// One wave32 per 16x16 output tile; 4 waves per block cover the 64 cols of a
// 16-row stripe. K=256 -> 64 WMMA steps of K=4.
// A frag (16x4): lanes 0-15 rows M=0..15 K={0,1}; lanes 16-31 K={2,3}.
// B frag (4x16): v0 holds K=0 (lanes 0-15) / K=2 (lanes 16-31); v1 K=1/K=3.
// C/D frag: VGPR v -> row (hi*8 + v), col = lane&15.
// ---------------------------------------------------------------------------
__global__ __launch_bounds__(128) void gat_gemm_wmma(
    const float* __restrict__ x, const float* __restrict__ W,
    float* __restrict__ h, int n_nodes)
{
  const int wave = threadIdx.x >> 5;   // 0..3 -> 16-col block
  const int lane = threadIdx.x & 31;
  const int row0 = blockIdx.x * 16;
  if (row0 >= n_nodes) return;
  const int n0 = wave * 16;

  const int r  = lane & 15;            // row (A) / col (B) within tile
  const int hi = lane >> 4;            // half-wave select
  const int kk = hi * 2;               // K sub-offset for this half

  const float* xrow = x + (size_t)(row0 + r) * IN_F;

  v8f c = {};
#pragma unroll 4
  for (int k0 = 0; k0 < IN_F; k0 += 4) {
    const float* ap = xrow + k0 + kk;
    v2f a;
    a.x = ap[0];                       // K = k0+kk
    a.y = ap[1];                       // K = k0+kk+1
    const float* bp = W + (size_t)(k0 + kk) * OUT_F + n0 + r;
    v2f b;
    b.x = bp[0];                       // K = k0+kk   , N = n0+r
    b.y = bp[OUT_F];                   // K = k0+kk+1 , N = n0+r
    c = __builtin_amdgcn_wmma_f32_16x16x4_f32(
        /*neg_a=*/false, a, /*neg_b=*/false, b,
        /*c_mod=*/(short)0, c, /*reuse_a=*/false, /*reuse_b=*/false);
  }

  float* outp = h + (size_t)(row0 + hi * 8) * OUT_F + n0 + r;
#pragma unroll
  for (int v = 0; v < 8; ++v)
    outp[(size_t)v * OUT_F] = c[v];
}

// ---------------------------------------------------------------------------
// Kernel 2: per-node attention scores s_src = h.a1, s_dst = h.a2
// ---------------------------------------------------------------------------
__global__ void gat_scores(const float* __restrict__ h,
                           const float* __restrict__ a,
                           float* __restrict__ s_src,
                           float* __restrict__ s_dst, int n_nodes)
{
  int i = blockIdx.x * blockDim.x + threadIdx.x;
  if (i >= n_nodes) return;
  const float* hr = h + (size_t)i * OUT_F;
  float s1 = 0.f, s2 = 0.f;
#pragma unroll 8
  for (int f = 0; f < OUT_F; ++f) {
    float hv = hr[f];
    s1 += hv * a[f];
    s2 += hv * a[OUT_F + f];
  }
  s_src[i] = s1;
  s_dst[i] = s2;
}

// ---------------------------------------------------------------------------
// Kernel 3: zero accumulators (graph-replay safe: runs every launch)
// ---------------------------------------------------------------------------
__global__ void gat_zero(float* __restrict__ out, float* __restrict__ rowsum,
                         int n_out, int n_nodes)
{
  int i = blockIdx.x * blockDim.x + threadIdx.x;
  if (i < n_out) out[i] = 0.f;
  if (i < n_nodes) rowsum[i] = 0.f;
}

// ---------------------------------------------------------------------------
// Kernel 4: edge scatter. One wave32 per edge; 2 features per lane.
// h gather and atomic scatter are 128B-contiguous per half-wave; rowsum
// updated once per edge (lane 0).
// ---------------------------------------------------------------------------
__global__ __launch_bounds__(256) void gat_edges(
    const int* __restrict__ ei, const float* __restrict__ h,
    const float* __restrict__ s_src, const float* __restrict__ s_dst,
    float* __restrict__ out, float* __restrict__ rowsum, int n_edges)
{
  long long t = (long long)blockIdx.x * blockDim.x + threadIdx.x;
  long long e = t >> 5;
  if (e >= n_edges) return;
  int lane = (int)(t & 31);

  int src = ei[e];
  int dst = ei[(size_t)n_edges + e];

  float score = s_src[src] + s_dst[dst];           // broadcast loads per wave
  float lr = score > 0.f ? score : GAT_ALPHA * score;
  float ee = expf(-lr);

  if (lane == 0) atomicAdd(rowsum + src, ee);

  const float* hv = h + (size_t)dst * OUT_F + lane * 2;
  float h0 = hv[0], h1 = hv[1];
  float* op = out + (size_t)src * OUT_F + lane * 2;
  atomicAdd(op,     ee * h0);
  atomicAdd(op + 1, ee * h1);
}

// ---------------------------------------------------------------------------
// Kernel 5: normalize + ELU, in place
// ---------------------------------------------------------------------------
__global__ void gat_finalize(float* __restrict__ out,
                             const float* __restrict__ rowsum, int n_total)
{
  int idx = blockIdx.x * blockDim.x + threadIdx.x;
  if (idx >= n_total) return;
  float v = out[idx] / rowsum[idx / OUT_F];
  out[idx] = v > 0.f ? v : expm1f(v);
}

// ---------------------------------------------------------------------------
extern "C" void kernel_launch(void* const* d_in, const int* in_sizes, int n_in,
                              void* d_out, int out_size, void* d_ws, size_t ws_size,
                              hipStream_t stream)
{
  const float* x  = (const float*)d_in[0];  // (N, 256)
  const float* W  = (const float*)d_in[1];  // (256, 64)
  const float* a  = (const float*)d_in[2];  // (1, 128)
  const int*   ei = (const int*)d_in[3];    // (2, E) int32

  float* out = (float*)d_out;               // (N, 64) f32

  const int n_nodes = in_sizes[0] / IN_F;   // 50000
  const int n_edges = in_sizes[3] / 2;      // 1,600,000
  const int n_out   = n_nodes * OUT_F;      // 3,200,000

  // workspace layout (floats): h | s_src | s_dst | rowsum  (~13.4 MB)
  float* h      = (float*)d_ws;
  float* s_src  = h + (size_t)n_nodes * OUT_F;
  float* s_dst  = s_src + n_nodes;
  float* rowsum = s_dst + n_nodes;

  gat_zero<<<(n_out + 255) / 256, 256, 0, stream>>>(out, rowsum, n_out, n_nodes);

  gat_gemm_wmma<<<(n_nodes + 15) / 16, 128, 0, stream>>>(x, W, h, n_nodes);

  gat_scores<<<(n_nodes + 255) / 256, 256, 0, stream>>>(h, a, s_src, s_dst, n_nodes);

  long long edge_threads = (long long)n_edges * 32;
  gat_edges<<<(unsigned)((edge_threads + 255) / 256), 256, 0, stream>>>(
      ei, h, s_src, s_dst, out, rowsum, n_edges);

  gat_finalize<<<(n_out + 255) / 256, 256, 0, stream>>>(out, rowsum, n_out);
}